// MVNDistribution_45226005626983
// MI455X (gfx1250) — compile-verified
//
#include <hip/hip_runtime.h>
#include <math.h>

// ---------------------------------------------------------------------------
// Problem constants (from reference)
// ---------------------------------------------------------------------------
#define NGAUSS 1024
#define KDIM   256
#define NOUT   9          // 3 means + 6 raw scales
#define DD     8
#define HH     64
#define WW     64
#define NPIX   (DD * HH * WW)   // 32768
#define EPSV   1e-06f

typedef __attribute__((ext_vector_type(2))) float v2f;
typedef __attribute__((ext_vector_type(8))) float v8f;

// ---------------------------------------------------------------------------
// Kernel 1: skinny GEMM via V_WMMA_F32_16X16X4_F32.
//   params[row][n] = rep[row,:] . Wc[n,:] + bc[n]
//   Wc rows 0..2 = mean_W, rows 3..8 = scale_W, rows 9..15 = 0 (padding).
// One wave (32 threads) per 16-row tile; 64 tiles.
//
// WMMA f32 16x16x4 operand layout (ISA 7.12.2):
//   A 16x4 : lanes 0-15 -> M=lane, VGPR{0,1}=K{0,1}; lanes 16-31 -> K{2,3}
//   B 4x16 : lanes 0-15 -> N=lane, VGPR{0,1}=K{0,1}; lanes 16-31 -> K{2,3}
//   C/D    : 8 VGPRs; N = lane&15; M = r + (lane>=16 ? 8 : 0)
// ---------------------------------------------------------------------------
__global__ __launch_bounds__(32)
void mvn_param_gemm_wmma(const float* __restrict__ rep,
                         const float* __restrict__ meanW,
                         const float* __restrict__ meanb,
                         const float* __restrict__ scaleW,
                         const float* __restrict__ scaleb,
                         float* __restrict__ params)
{
    __shared__ float wlds[16 * KDIM];   // zero-padded combined weights, 16 KB

    const int lane = threadIdx.x;       // 0..31 (wave32)
    const int tile = blockIdx.x;        // 0..63
    const int half = lane >> 4;         // which K-pair this lane holds
    const int l    = lane & 15;         // M index (A) / N index (B, D)

    // Stage combined, zero-padded weight block into LDS (row n, col k).
    for (int idx = lane; idx < 16 * KDIM; idx += 32) {
        const int n = idx >> 8;         // 0..15
        const int k = idx & (KDIM - 1);
        float v = 0.0f;
        if (n < 3)      v = meanW[n * KDIM + k];
        else if (n < 9) v = scaleW[(n - 3) * KDIM + k];
        wlds[idx] = v;
    }
    __syncthreads();

    const float* __restrict__ arow = rep + (size_t)(tile * 16 + l) * KDIM + half * 2;
    const float* __restrict__ brow = wlds + l * KDIM + half * 2;

    v8f acc = {};
    #pragma unroll 4
    for (int k = 0; k < KDIM; k += 4) {
        v2f a; a.x = arow[k]; a.y = arow[k + 1];
        v2f b; b.x = brow[k]; b.y = brow[k + 1];
        // D = A*B + C, fp32 WMMA (16x16x4), 8 args:
        // (neg_a, A, neg_b, B, c_mod, C, reuse_a, reuse_b)
        acc = __builtin_amdgcn_wmma_f32_16x16x4_f32(
            false, a, false, b, (short)0, acc, false, false);
    }

    // Write the 9 live columns with bias.
    if (l < NOUT) {
        const float bias = (l < 3) ? meanb[l] : scaleb[l - 3];
        #pragma unroll
        for (int r = 0; r < 8; ++r) {
            const int row = tile * 16 + half * 8 + r;
            params[row * NOUT + l] = acc[r] + bias;
        }
    }
}

// ---------------------------------------------------------------------------
// Kernel 2: per-Gaussian profile (softmax over 32768 pixels).
// One block per Gaussian; maha recomputed per pass (compute is free vs the
// mandatory 128 MB store). log_det and the 1.5*log(2*pi) constant cancel in
// the normalized profile, so only maha matters:
//   profile_j = exp(-0.5*(maha_j - min_maha)) / (sum + 1e-10)
// ---------------------------------------------------------------------------
__device__ __forceinline__ float softplusf(float x) {
    return fmaxf(x, 0.0f) + log1pf(expf(-fabsf(x)));
}

#define TPB2 256

__global__ __launch_bounds__(TPB2)
void mvn_profile_kernel(const float* __restrict__ params,
                        float* __restrict__ out)
{
    const int row = blockIdx.x;
    const int tid = threadIdx.x;

    // --- per-row parameters (uniform across block) ---
    const float* p = params + row * NOUT;
    const float m0 = p[0], m1 = p[1], m2 = p[2];
    const float s0 = softplusf(p[3]) + EPSV;
    const float s1 = softplusf(p[4]) + EPSV;
    const float s2 = softplusf(p[5]) + EPSV;
    const float s3 = softplusf(p[6]) + EPSV;
    const float s4 = softplusf(p[7]) + EPSV;
    const float s5 = softplusf(p[8]) + EPSV;
    const float L00 = softplusf(s0);
    const float L10 = s1;
    const float L11 = softplusf(s2);
    const float L20 = s3;
    const float L21 = s4;
    const float L22 = softplusf(s5);
    const float i00 = 1.0f / L00, i11 = 1.0f / L11, i22 = 1.0f / L22;

    // maha evaluator: pid -> (x,y,z) pixel, whitened squared norm
    auto maha_at = [&](int pid) -> float {
        const float px = (float)(pid & 63)        - 31.5f;
        const float py = (float)((pid >> 6) & 63) - 31.5f;
        const float pz = (float)(pid >> 12)       - 3.5f;
        const float d0 = px - m0;
        const float d1 = py - m1;
        const float d2 = pz - m2;
        const float y0 = d0 * i00;
        const float y1 = (d1 - L10 * y0) * i11;
        const float y2 = (d2 - L20 * y0 - L21 * y1) * i22;
        return y0 * y0 + y1 * y1 + y2 * y2;
    };

    __shared__ float red[TPB2 / 32];

    // ---- pass 1: min(maha) over the row ----
    float lmin = 3.4e38f;
    for (int pid = tid; pid < NPIX; pid += TPB2)
        lmin = fminf(lmin, maha_at(pid));
    #pragma unroll
    for (int off = 16; off > 0; off >>= 1)
        lmin = fminf(lmin, __shfl_xor(lmin, off, 32));
    if ((tid & 31) == 0) red[tid >> 5] = lmin;
    __syncthreads();
    if (tid < 32) {
        float v = (tid < TPB2 / 32) ? red[tid] : 3.4e38f;
        #pragma unroll
        for (int off = 16; off > 0; off >>= 1)
            v = fminf(v, __shfl_xor(v, off, 32));
        if (tid == 0) red[0] = v;
    }
    __syncthreads();
    const float minm = red[0];
    __syncthreads();

    // ---- pass 2: sum of exp(-0.5*(maha - minm)) ----
    float lsum = 0.0f;
    for (int pid = tid; pid < NPIX; pid += TPB2)
        lsum += expf(-0.5f * (maha_at(pid) - minm));
    #pragma unroll
    for (int off = 16; off > 0; off >>= 1)
        lsum += __shfl_xor(lsum, off, 32);
    if ((tid & 31) == 0) red[tid >> 5] = lsum;
    __syncthreads();
    if (tid < 32) {
        float v = (tid < TPB2 / 32) ? red[tid] : 0.0f;
        #pragma unroll
        for (int off = 16; off > 0; off >>= 1)
            v += __shfl_xor(v, off, 32);
        if (tid == 0) red[0] = v;
    }
    __syncthreads();
    const float inv = 1.0f / (red[0] + 1e-10f);

    // ---- pass 3: write normalized profile (the only big store) ----
    float* __restrict__ orow = out + (size_t)row * NPIX;
    for (int pid = tid; pid < NPIX; pid += TPB2)
        orow[pid] = expf(-0.5f * (maha_at(pid) - minm)) * inv;
}

// ---------------------------------------------------------------------------
// Launch
// ---------------------------------------------------------------------------
extern "C" void kernel_launch(void* const* d_in, const int* in_sizes, int n_in,
                              void* d_out, int out_size, void* d_ws, size_t ws_size,
                              hipStream_t stream)
{
    const float* rep    = (const float*)d_in[0];   // 1024*256
    const float* meanW  = (const float*)d_in[1];   // 3*256
    const float* meanb  = (const float*)d_in[2];   // 3
    const float* scaleW = (const float*)d_in[3];   // 6*256
    const float* scaleb = (const float*)d_in[4];   // 6
    float*       out    = (float*)d_out;           // 1024*32768
    float*       params = (float*)d_ws;            // 1024*9 scratch

    mvn_param_gemm_wmma<<<NGAUSS / 16, 32, 0, stream>>>(
        rep, meanW, meanb, scaleW, scaleb, params);

    mvn_profile_kernel<<<NGAUSS, TPB2, 0, stream>>>(params, out);
}